// HybridLastHopGCNWrapper_62560493634015
// MI455X (gfx1250) — compile-verified
//
#include <hip/hip_runtime.h>

typedef __attribute__((ext_vector_type(2))) float v2f;
typedef __attribute__((ext_vector_type(8))) float v8f;

// ---------------- utility kernels ----------------

__global__ void k_init(float* __restrict__ deg, int n, int* __restrict__ maxd) {
  int i = blockIdx.x * blockDim.x + threadIdx.x;
  if (i < n) deg[i] = 1.0f;          // self-loop contributes 1 to in-degree
  if (i == 0) *maxd = 0;             // hop depths are >= 0
}

__global__ void k_max_depth(const int* __restrict__ hop, int n, int* __restrict__ maxd) {
  int local = 0;
  for (int i = blockIdx.x * blockDim.x + threadIdx.x; i < n; i += gridDim.x * blockDim.x)
    local = max(local, hop[i]);
  #pragma unroll
  for (int off = 16; off > 0; off >>= 1)
    local = max(local, __shfl_down(local, off, 32));
  if ((threadIdx.x & 31) == 0) atomicMax(maxd, local);
}

__global__ void k_deg_edges(const int* __restrict__ dst, int e, float* __restrict__ deg) {
  int i = blockIdx.x * blockDim.x + threadIdx.x;
  if (i < e) atomicAdd(&deg[dst[i]], 1.0f);
}

__global__ void k_dinv(float* __restrict__ d, int n) {
  int i = blockIdx.x * blockDim.x + threadIdx.x;
  if (i < n) d[i] = rsqrtf(d[i]);    // deg >= 1 always (self-loop)
}

// ---------------- WMMA GEMM: Y[M,NT*16] = X[M,KDIM] @ W[KDIM,NT*16] ----------------
// 8 waves/block, each wave computes a 16-row strip across all NT*16 columns.
// W is staged in LDS with a +1 dword pad per row to kill 2-way bank conflicts
// between lanes L and L+16 (row stride 65/33 dwords instead of 64/32).
// Optional fused input mask: rows with hop[row]==*maxd_ptr contribute zeros
// (applied as a 0/1 scale on the A fragment so EXEC stays all-ones for WMMA).

template<int KDIM, int NT>
__global__ __launch_bounds__(256) void k_gemm_wmma(
    const float* __restrict__ X, const float* __restrict__ W,
    const float* __restrict__ bias, float* __restrict__ Y, int M,
    const int* __restrict__ hop, const int* __restrict__ maxd_ptr, int relu)
{
  constexpr int NCOL = NT * 16;
  constexpr int LSTR = NCOL + 1;
  __shared__ float sW[KDIM * LSTR];
  for (int i = threadIdx.x; i < KDIM * NCOL; i += 256) {
    int r = i / NCOL, c = i - r * NCOL;
    sW[r * LSTR + c] = W[i];
  }
  __syncthreads();

  const int wave = threadIdx.x >> 5;
  const int lane = threadIdx.x & 31;
  const int l16  = lane & 15;
  const int lhi  = lane >> 4;                 // 0: lanes 0-15, 1: lanes 16-31
  const int m0   = blockIdx.x * 128 + wave * 16;
  const int row  = m0 + l16;
  const bool rvalid = row < M;

  float scale = rvalid ? 1.0f : 0.0f;
  if (hop && rvalid && hop[row] == *maxd_ptr) scale = 0.0f;
  const float* xrow = X + (size_t)(rvalid ? row : 0) * KDIM;

  v8f acc[NT] = {};
  #pragma unroll
  for (int k = 0; k < KDIM; k += 4) {
    // A fragment (ISA 32-bit 16x4 layout): v0=K{k,k+2}, v1=K{k+1,k+3}
    const int ka = k + lhi * 2;
    v2f a = *(const v2f*)(xrow + ka);         // 8B-aligned: ka is even
    a *= scale;
    #pragma unroll
    for (int t = 0; t < NT; ++t) {
      const int col = t * 16 + l16;
      v2f b;
      b.x = sW[(ka    ) * LSTR + col];        // B row-striped: K=ka / ka+1
      b.y = sW[(ka + 1) * LSTR + col];
      acc[t] = __builtin_amdgcn_wmma_f32_16x16x4_f32(
          false, a, false, b, (short)0, acc[t], false, false);
    }
  }

  // C layout: VGPR r holds M = m0 + r + 8*lhi, N = t*16 + l16
  #pragma unroll
  for (int t = 0; t < NT; ++t) {
    const int n  = t * 16 + l16;
    const float bv = bias ? bias[n] : 0.0f;
    #pragma unroll
    for (int r = 0; r < 8; ++r) {
      const int rr = m0 + r + 8 * lhi;
      if (rr < M) {
        float v = acc[t][r] + bv;
        if (relu) v = fmaxf(v, 0.0f);
        Y[(size_t)rr * NCOL + n] = v;
      }
    }
  }
}

// ---------------- aggregation ----------------

// agg[n] = xw[n] * dinv[n]^2   (self-loop term; also zero-initializes agg)
__global__ void k_self_init(const float* __restrict__ xw, const float* __restrict__ dinv,
                            float* __restrict__ agg, int n) {
  int t = blockIdx.x * blockDim.x + threadIdx.x;
  int node = t >> 4;
  if (node >= n) return;
  int c = (t & 15) * 4;
  float w = dinv[node]; w *= w;
  float4 v = *(const float4*)(xw + (size_t)node * 64 + c);
  *(float4*)(agg + (size_t)node * 64 + c) = make_float4(v.x * w, v.y * w, v.z * w, v.w * w);
}

// agg[dst] += xw[src] * dinv[src]*dinv[dst]; 16 lanes/edge, float4 per lane.
// Working set (xw + agg ~= 51 MB) is L2-resident on MI455X (192 MB L2).
__global__ void k_edge_agg(const int* __restrict__ src, const int* __restrict__ dst,
                           const float* __restrict__ dinv, const float* __restrict__ xw,
                           float* __restrict__ agg, int e) {
  int t = blockIdx.x * blockDim.x + threadIdx.x;
  int ei = t >> 4;
  if (ei >= e) return;
  int c = (t & 15) * 4;
  int s = src[ei], d = dst[ei];
  float w = dinv[s] * dinv[d];
  float4 v = *(const float4*)(xw + (size_t)s * 64 + c);
  float* o = agg + (size_t)d * 64 + c;
  atomicAdd(o + 0, v.x * w);
  atomicAdd(o + 1, v.y * w);
  atomicAdd(o + 2, v.z * w);
  atomicAdd(o + 3, v.w * w);
}

// out = relu( (frontier ? rep : agg) + bias ), frontier = hop == maxd-1.
// rep==nullptr disables the frontier select (layer-2 epilogue).
__global__ void k_post(const float* __restrict__ agg, const float* __restrict__ rep,
                       const int* __restrict__ hop, const int* __restrict__ maxd_ptr,
                       const float* __restrict__ bias, float* __restrict__ out, int n) {
  int t = blockIdx.x * blockDim.x + threadIdx.x;
  int node = t >> 4;
  if (node >= n) return;
  int c = (t & 15) * 4;
  const float* srcp = agg;
  if (rep && hop[node] == (*maxd_ptr) - 1) srcp = rep;
  float4 v = *(const float4*)(srcp + (size_t)node * 64 + c);
  float4 b = *(const float4*)(bias + c);
  float4 r;
  r.x = fmaxf(v.x + b.x, 0.0f);
  r.y = fmaxf(v.y + b.y, 0.0f);
  r.z = fmaxf(v.z + b.z, 0.0f);
  r.w = fmaxf(v.w + b.w, 0.0f);
  *(float4*)(out + (size_t)node * 64 + c) = r;
}

// ---------------- launch ----------------

extern "C" void kernel_launch(void* const* d_in, const int* in_sizes, int n_in,
                              void* d_out, int out_size, void* d_ws, size_t ws_size,
                              hipStream_t stream) {
  const float* x    = (const float*)d_in[0];
  const int*   eidx = (const int*)  d_in[1];
  const int*   hop  = (const int*)  d_in[2];
  const float* pre  = (const float*)d_in[3];
  const float* W1   = (const float*)d_in[4];
  const float* b1   = (const float*)d_in[5];
  const float* W2   = (const float*)d_in[6];
  const float* b2   = (const float*)d_in[7];
  const float* Wc   = (const float*)d_in[8];
  const float* bc   = (const float*)d_in[9];

  const int N = in_sizes[0] / 64;       // 100000
  const int E = in_sizes[1] / 2;        // 1200000
  const int* srcI = eidx;
  const int* dstI = eidx + E;

  char* ws = (char*)d_ws;
  int*   maxd = (int*)ws;
  float* dinv = (float*)(ws + 256);
  size_t dOff = 256 + (((size_t)N * sizeof(float) + 255) & ~(size_t)255);
  size_t nodeBytes = (size_t)N * 64 * sizeof(float);
  float* bufA = (float*)(ws + dOff);                     // xw1 -> hidden -> h2
  float* bufB = (float*)(ws + dOff + nodeBytes);         // rep -> xw2
  float* bufC = (float*)(ws + dOff + 2 * nodeBytes);     // agg1 -> agg2

  dim3 blk(256);
  int gN   = (N + 255) / 256;
  int gE   = (E + 255) / 256;
  int gN16 = (int)(((size_t)N * 16 + 255) / 256);
  int gE16 = (int)(((size_t)E * 16 + 255) / 256);
  int gG   = (N + 127) / 128;

  // norms + max depth
  k_init     <<<gN,  blk, 0, stream>>>(dinv, N, maxd);
  k_max_depth<<<256, blk, 0, stream>>>(hop, N, maxd);
  k_deg_edges<<<gE,  blk, 0, stream>>>(dstI, E, dinv);
  k_dinv     <<<gN,  blk, 0, stream>>>(dinv, N);

  // layer 1: xw1 = (deepest? 0 : x) @ W1 ; rep = preagg @ W1
  k_gemm_wmma<64, 4><<<gG, blk, 0, stream>>>(x,   W1, nullptr, bufA, N, hop, maxd, 0);
  k_gemm_wmma<64, 4><<<gG, blk, 0, stream>>>(pre, W1, nullptr, bufB, N, nullptr, nullptr, 0);
  k_self_init<<<gN16, blk, 0, stream>>>(bufA, dinv, bufC, N);
  k_edge_agg <<<gE16, blk, 0, stream>>>(srcI, dstI, dinv, bufA, bufC, E);
  k_post     <<<gN16, blk, 0, stream>>>(bufC, bufB, hop, maxd, b1, bufA, N);   // hidden

  // layer 2
  k_gemm_wmma<64, 4><<<gG, blk, 0, stream>>>(bufA, W2, nullptr, bufB, N, nullptr, nullptr, 0);
  k_self_init<<<gN16, blk, 0, stream>>>(bufB, dinv, bufC, N);
  k_edge_agg <<<gE16, blk, 0, stream>>>(srcI, dstI, dinv, bufB, bufC, E);
  k_post     <<<gN16, blk, 0, stream>>>(bufC, nullptr, nullptr, nullptr, b2, bufA, N); // h2

  // classifier: d_out = h2 @ Wc + bc   (N x 32)
  k_gemm_wmma<64, 2><<<gG, blk, 0, stream>>>(bufA, Wc, bc, (float*)d_out, N,
                                             nullptr, nullptr, 0);
}